// SemanticConv_62895501082649
// MI455X (gfx1250) — compile-verified
//
#include <hip/hip_runtime.h>
#include <math.h>

// Problem constants (from reference)
#define BB 4
#define DD 64
#define NN 4096
#define OO 128
#define KNN 16
#define NEG_SLOPE 0.01f
#define EPS 1e-5f

typedef float v2f __attribute__((ext_vector_type(2)));
typedef float v8f __attribute__((ext_vector_type(8)));

// ---------------------------------------------------------------------------
// WMMA helpers: V_WMMA_F32_16X16X4_F32, wave32.
//   A (16x4 f32): lanes 0-15 hold M=lane, v0=K0,v1=K1; lanes 16-31: v0=K2,v1=K3
//   B (4x16 f32): lanes 0-15 hold N=lane, v0=K0,v1=K1; lanes 16-31: v0=K2,v1=K3
//   C/D (16x16):  vgpr r: lanes 0-15 -> (M=r, N=lane); lanes 16-31 -> (M=r+8)
// ---------------------------------------------------------------------------
__device__ __forceinline__ v8f wmma4(v2f a, v2f b, v8f c) {
  return __builtin_amdgcn_wmma_f32_16x16x4_f32(false, a, false, b, (short)0, c,
                                               false, false);
}

// A row-major: A[m*lda + k]
__device__ __forceinline__ v2f ld_a_rm(const float* A, int lda, int m0, int k0,
                                       int lane) {
  int m = lane & 15, kk = (lane >> 4) << 1;
  v2f r;
  r.x = A[(size_t)(m0 + m) * lda + k0 + kk];
  r.y = A[(size_t)(m0 + m) * lda + k0 + kk + 1];
  return r;
}
// A row-major with K-bound guard (zero-pad)
__device__ __forceinline__ v2f ld_a_rm_g(const float* A, int lda, int m0,
                                         int k0, int lane, int klim) {
  int m = lane & 15, kk = (lane >> 4) << 1;
  int c0 = k0 + kk, c1 = c0 + 1;
  v2f r;
  r.x = (c0 < klim) ? A[(size_t)(m0 + m) * lda + c0] : 0.f;
  r.y = (c1 < klim) ? A[(size_t)(m0 + m) * lda + c1] : 0.f;
  return r;
}
// A column-major: A[k*ldk + m]  (for f stored [d][n], m-index = n)
__device__ __forceinline__ v2f ld_a_cm(const float* A, int ldk, int m0, int k0,
                                       int lane) {
  int m = lane & 15, kk = (lane >> 4) << 1;
  v2f r;
  r.x = A[(size_t)(k0 + kk) * ldk + m0 + m];
  r.y = A[(size_t)(k0 + kk + 1) * ldk + m0 + m];
  return r;
}
// B row-major: B[k*ldb + n]
__device__ __forceinline__ v2f ld_b_rm(const float* Bp, int ldb, int k0, int n0,
                                       int lane) {
  int n = lane & 15, kk = (lane >> 4) << 1;
  v2f r;
  r.x = Bp[(size_t)(k0 + kk) * ldb + n0 + n];
  r.y = Bp[(size_t)(k0 + kk + 1) * ldb + n0 + n];
  return r;
}

// reductions within each 16-lane half of a wave32
__device__ __forceinline__ float half_max(float v) {
  v = fmaxf(v, __shfl_xor(v, 1, 32));
  v = fmaxf(v, __shfl_xor(v, 2, 32));
  v = fmaxf(v, __shfl_xor(v, 4, 32));
  v = fmaxf(v, __shfl_xor(v, 8, 32));
  return v;
}
__device__ __forceinline__ float half_sum(float v) {
  v += __shfl_xor(v, 1, 32);
  v += __shfl_xor(v, 2, 32);
  v += __shfl_xor(v, 4, 32);
  v += __shfl_xor(v, 8, 32);
  return v;
}

// ---------------------------------------------------------------------------
// 1) xx[b,n] = sum_d f^2
// ---------------------------------------------------------------------------
__global__ void k_sq(const float* __restrict__ f, float* __restrict__ xx) {
  int i = blockIdx.x * blockDim.x + threadIdx.x;  // 0..B*N
  if (i >= BB * NN) return;
  int b = i / NN, n = i % NN;
  const float* F = f + (size_t)b * DD * NN;
  float s = 0.f;
#pragma unroll 8
  for (int d = 0; d < DD; ++d) {
    float v = F[(size_t)d * NN + n];
    s += v * v;
  }
  xx[i] = s;
}

// ---------------------------------------------------------------------------
// 2) q/k/v projections: out[b,o,n] = sum_d W[o,d] f[b,d,n] + bias[o]
//    grid (B*N/16, 3), block 128 (4 waves = 4 o-tiles of 16)
// ---------------------------------------------------------------------------
__global__ void k_proj(const float* __restrict__ f, const float* __restrict__ wq,
                       const float* __restrict__ bq, const float* __restrict__ wk,
                       const float* __restrict__ bk, const float* __restrict__ wv,
                       const float* __restrict__ bv, float* __restrict__ q,
                       float* __restrict__ k, float* __restrict__ v) {
  int job = blockIdx.x;
  int b = job >> 8;
  int n0 = (job & 255) << 4;
  int lane = threadIdx.x & 31;
  int w = threadIdx.x >> 5;  // o-tile 0..3
  const float* W;
  const float* bias;
  float* out;
  if (blockIdx.y == 0) { W = wq; bias = bq; out = q; }
  else if (blockIdx.y == 1) { W = wk; bias = bk; out = k; }
  else { W = wv; bias = bv; out = v; }
  const float* F = f + (size_t)b * DD * NN;
  float* Ob = out + (size_t)b * DD * NN;
  v8f acc = {};
#pragma unroll
  for (int kc = 0; kc < DD / 4; ++kc) {
    v2f a = ld_a_rm(W, DD, w * 16, kc * 4, lane);
    v2f bt = ld_b_rm(F, NN, kc * 4, n0, lane);
    acc = wmma4(a, bt, acc);
  }
  int h = lane >> 4, c = lane & 15;
#pragma unroll
  for (int r = 0; r < 8; ++r) {
    int o = w * 16 + r + 8 * h;
    Ob[(size_t)o * NN + n0 + c] = acc[r] + bias[o];
  }
}

// ---------------------------------------------------------------------------
// 3) pd Gram + top-16 indices per row.
//    block 128 (4 waves), 1 block per (b, 16-row m-tile) -> 1024 blocks
// ---------------------------------------------------------------------------
__global__ void k_topk(const float* __restrict__ f, const float* __restrict__ xx,
                       int* __restrict__ topk) {
  __shared__ float pdbuf[16 * 64];
  __shared__ float candv[16 * 128];
  __shared__ int candi[16 * 128];
  int blk = blockIdx.x;
  int b = blk >> 8;
  int m0 = (blk & 255) << 4;
  const float* F = f + (size_t)b * DD * NN;
  const float* XX = xx + (size_t)b * NN;
  int tid = threadIdx.x, lane = tid & 31, w = tid >> 5;
  int row = tid >> 3, s8 = tid & 7;
  int h = lane >> 4, cL = lane & 15;

  float xm[8];
#pragma unroll
  for (int r = 0; r < 8; ++r) xm[r] = XX[m0 + r + 8 * h];

  float tv[16];
  int ti[16];
#pragma unroll
  for (int i = 0; i < 16; ++i) { tv[i] = -3.0e38f; ti[i] = 0; }
  float minv = -3.0e38f;
  int minp = 0;

  for (int st = 0; st < 64; ++st) {
    int n0 = (st * 4 + w) * 16;
    if (st < 63) __builtin_prefetch(&F[(st + 1) * 64 + w * 16], 0, 1);
    v8f acc = {};
#pragma unroll
    for (int kc = 0; kc < DD / 4; ++kc) {
      v2f a = ld_a_cm(F, NN, m0, kc * 4, lane);
      v2f bt = ld_b_rm(F, NN, kc * 4, n0, lane);
      acc = wmma4(a, bt, acc);
    }
    float xn = XX[n0 + cL];
#pragma unroll
    for (int r = 0; r < 8; ++r) {
      float pd = 2.0f * acc[r] - xm[r] - xn;
      pdbuf[(r + 8 * h) * 64 + w * 16 + cL] = pd;
    }
    __syncthreads();
#pragma unroll
    for (int j = 0; j < 8; ++j) {
      int cc = s8 + 8 * j;
      float vv = pdbuf[row * 64 + cc];
      if (vv > minv) {
        tv[minp] = vv;
        ti[minp] = st * 64 + cc;
        minv = tv[0];
        minp = 0;
#pragma unroll
        for (int qq = 1; qq < 16; ++qq)
          if (tv[qq] < minv) { minv = tv[qq]; minp = qq; }
      }
    }
    __syncthreads();
  }
#pragma unroll
  for (int i = 0; i < 16; ++i) {
    candv[row * 128 + s8 * 16 + i] = tv[i];
    candi[row * 128 + s8 * 16 + i] = ti[i];
  }
  __syncthreads();
  if (tid < 16) {
    int r = tid;
    for (int i = 0; i < 16; ++i) {
      float best = -3.3e38f;
      int bp = 0;
      for (int qq = 0; qq < 128; ++qq) {
        float vv = candv[r * 128 + qq];
        if (vv > best) { best = vv; bp = qq; }
      }
      topk[((size_t)b * NN + m0 + r) * KNN + i] = candi[r * 128 + bp];
      candv[r * 128 + bp] = -3.4e38f;
    }
  }
}

// ---------------------------------------------------------------------------
// 4) attention: fgt[b,d,m] = sum_n softmax(q^T k / 8)[m,n] v[d,n]
//    block 256 = 8 independent waves, each owns a 16-row m-tile. 128 blocks.
// ---------------------------------------------------------------------------
__global__ void k_attn(const float* __restrict__ q, const float* __restrict__ km,
                       const float* __restrict__ v, float* __restrict__ fgt) {
  __shared__ float Pbuf[8][256];
  int wav = threadIdx.x >> 5;
  int wgm = blockIdx.x * 8 + wav;  // 0..1023
  int b = wgm >> 8;
  int m0 = (wgm & 255) << 4;
  int lane = threadIdx.x & 31, h = lane >> 4, cL = lane & 15;
  const float* Q = q + (size_t)b * DD * NN;
  const float* Km = km + (size_t)b * DD * NN;
  const float* V = v + (size_t)b * DD * NN;
  float* FG = fgt + (size_t)b * DD * NN;
  float* myP = Pbuf[wav];
  const float scale = 0.125f;  // 1/sqrt(64)

  // pass 1: row maxima
  float rmax[8];
#pragma unroll
  for (int r = 0; r < 8; ++r) rmax[r] = -3.0e38f;
  for (int nt = 0; nt < 256; ++nt) {
    v8f acc = {};
#pragma unroll
    for (int kc = 0; kc < DD / 4; ++kc) {
      v2f a = ld_a_cm(Q, NN, m0, kc * 4, lane);
      v2f bt = ld_b_rm(Km, NN, kc * 4, nt * 16, lane);
      acc = wmma4(a, bt, acc);
    }
#pragma unroll
    for (int r = 0; r < 8; ++r) rmax[r] = fmaxf(rmax[r], acc[r] * scale);
  }
#pragma unroll
  for (int r = 0; r < 8; ++r) rmax[r] = half_max(rmax[r]);

  // pass 2: exp, Z, and P @ V^T accumulation (4 d-tiles of 16)
  float zs[8] = {};
  v8f fac[4] = {};
  for (int nt = 0; nt < 256; ++nt) {
    v8f acc = {};
#pragma unroll
    for (int kc = 0; kc < DD / 4; ++kc) {
      v2f a = ld_a_cm(Q, NN, m0, kc * 4, lane);
      v2f bt = ld_b_rm(Km, NN, kc * 4, nt * 16, lane);
      acc = wmma4(a, bt, acc);
    }
#pragma unroll
    for (int r = 0; r < 8; ++r) {
      float p = __expf(acc[r] * scale - rmax[r]);
      zs[r] += p;
      myP[(r + 8 * h) * 16 + cL] = p;  // in-order wave-private LDS
    }
    int kk = (lane >> 4) << 1;
#pragma unroll
    for (int t = 0; t < 4; ++t) {
#pragma unroll
      for (int kc = 0; kc < 4; ++kc) {
        v2f a;  // A = P tile (row-major 16x16 in LDS), K = n-in-tile
        a.x = myP[cL * 16 + kc * 4 + kk];
        a.y = myP[cL * 16 + kc * 4 + kk + 1];
        v2f bt;  // B[k_n, j_d] = V[(d0+j)*N + n], column-major in memory
        bt.x = V[(size_t)(t * 16 + cL) * NN + nt * 16 + kc * 4 + kk];
        bt.y = V[(size_t)(t * 16 + cL) * NN + nt * 16 + kc * 4 + kk + 1];
        fac[t] = wmma4(a, bt, fac[t]);
      }
    }
  }
#pragma unroll
  for (int r = 0; r < 8; ++r) zs[r] = half_sum(zs[r]);
#pragma unroll
  for (int t = 0; t < 4; ++t)
#pragma unroll
    for (int r = 0; r < 8; ++r) {
      int m = m0 + r + 8 * h, d = t * 16 + cL;
      FG[(size_t)d * NN + m] = fac[t][r] / zs[r];
    }
}

// ---------------------------------------------------------------------------
// 5) intra: per point build feat(129x16) in LDS, 128x129 GEMM via WMMA,
//    max over k to intra_pre; per-channel (sum,sumsq) partials for BN.
//    block 256 (8 waves = 8 o-tiles), 16 points per block, 1024 blocks.
// ---------------------------------------------------------------------------
__global__ void k_intra(const float* __restrict__ f,
                        const float* __restrict__ w_local,
                        const int* __restrict__ topk,
                        float* __restrict__ intra_pre,
                        float* __restrict__ part) {
  __shared__ float feat[132 * 16];
  __shared__ float chacc[256];
  __shared__ int jidx[16];
  int tid = threadIdx.x, lane = tid & 31, w = tid >> 5;
  int h = lane >> 4, cL = lane & 15;
  if (tid < 256) chacc[tid] = 0.f;
  __syncthreads();

  for (int p = 0; p < 16; ++p) {
    int pid = blockIdx.x * 16 + p;
    int b = pid >> 12, n = pid & (NN - 1);
    const float* F = f + (size_t)b * DD * NN;
    if (tid < 16) jidx[tid] = topk[(size_t)pid * KNN + tid];
    __syncthreads();
    for (int e = tid; e < 128 * 16; e += 256) {
      int cc = e >> 4, kx = e & 15;
      feat[cc * 16 + kx] = (cc < DD) ? F[(size_t)cc * NN + n]
                                     : F[(size_t)(cc - DD) * NN + jidx[kx]];
    }
    if (tid < 16) {
      int j = jidx[tid];
      float s = 0.f;
#pragma unroll 8
      for (int d = 0; d < DD; ++d) {
        float df = F[(size_t)d * NN + j] - F[(size_t)d * NN + n];
        s += df * df;
      }
      feat[128 * 16 + tid] = sqrtf(s);
    } else if (tid < 64) {
      feat[(129 + ((tid - 16) >> 4)) * 16 + (tid & 15)] = 0.f;  // pad rows
    }
    __syncthreads();

    int o0 = w * 16;
    v8f acc = {};
    int kk = (lane >> 4) << 1;
#pragma unroll
    for (int ch = 0; ch < 33; ++ch) {
      int k0 = ch * 4;
      v2f a = ld_a_rm_g(w_local, 129, o0, k0, lane, 129);
      v2f bt;
      bt.x = feat[(k0 + kk) * 16 + cL];
      bt.y = feat[(k0 + kk + 1) * 16 + cL];
      acc = wmma4(a, bt, acc);
    }
#pragma unroll
    for (int r = 0; r < 8; ++r) {
      float y = acc[r];
      float rm = half_max(y);
      float rs = half_sum(y);
      float rq = half_sum(y * y);
      if (cL == r) {
        int o = o0 + r + 8 * h;
        intra_pre[((size_t)b * OO + o) * NN + n] = rm;
        chacc[o * 2] += rs;
        chacc[o * 2 + 1] += rq;
      }
    }
    __syncthreads();
  }
  if (tid < 128) {
    part[((size_t)blockIdx.x * OO + tid) * 2] = chacc[tid * 2];
    part[((size_t)blockIdx.x * OO + tid) * 2 + 1] = chacc[tid * 2 + 1];
  }
}

// ---------------------------------------------------------------------------
// 6) generic channel GEMM: pre[b,o,n] = sum_c W[o,c] src[c][b, :, n]
//    src splits at `split` between s1 (ch1 channels) and s2 (ch2 channels).
//    Writes pre + per-block (sum,sumsq) partials. block 256 (8 o-tiles).
// ---------------------------------------------------------------------------
__global__ void k_cgemm(const float* __restrict__ W, int Kdim,
                        const float* __restrict__ s1, int ch1,
                        const float* __restrict__ s2, int ch2, int split,
                        float* __restrict__ pre, float* __restrict__ part) {
  int b = blockIdx.x >> 8;
  int n0 = (blockIdx.x & 255) << 4;
  int lane = threadIdx.x & 31, w = threadIdx.x >> 5;
  int h = lane >> 4, cL = lane & 15;
  int o0 = w * 16;
  v8f acc = {};
  for (int kc = 0; kc < Kdim / 4; ++kc) {
    int k0 = kc * 4;
    v2f a = ld_a_rm(W, Kdim, o0, k0, lane);
    const float* S;
    int cb;
    if (k0 < split) { S = s1 + (size_t)b * ch1 * NN; cb = k0; }
    else            { S = s2 + (size_t)b * ch2 * NN; cb = k0 - split; }
    v2f bt = ld_b_rm(S, NN, cb, n0, lane);
    acc = wmma4(a, bt, acc);
  }
#pragma unroll
  for (int r = 0; r < 8; ++r) {
    float y = acc[r];
    int o = o0 + r + 8 * h;
    pre[((size_t)b * OO + o) * NN + n0 + cL] = y;
    float rs = half_sum(y);
    float rq = half_sum(y * y);
    if (cL == r) {
      part[((size_t)blockIdx.x * OO + o) * 2] = rs;
      part[((size_t)blockIdx.x * OO + o) * 2 + 1] = rq;
    }
  }
}

// ---------------------------------------------------------------------------
// 7) fixed-order stats reduction (deterministic): mean + inv_std per channel
// ---------------------------------------------------------------------------
__global__ void k_redstats(const float* __restrict__ part, int P, float cnt,
                           float* __restrict__ stats) {
  int o = threadIdx.x;
  if (o >= OO) return;
  float s1 = 0.f, s2 = 0.f;
  for (int pblk = 0; pblk < P; ++pblk) {
    s1 += part[((size_t)pblk * OO + o) * 2];
    s2 += part[((size_t)pblk * OO + o) * 2 + 1];
  }
  float mean = s1 / cnt;
  float var = s2 / cnt - mean * mean;
  stats[o * 2] = mean;
  stats[o * 2 + 1] = rsqrtf(var + EPS);
}

// ---------------------------------------------------------------------------
// 8) BN + leaky-relu finalize (elementwise, may be in-place)
// ---------------------------------------------------------------------------
__global__ void k_finalize(const float* __restrict__ pre,
                           const float* __restrict__ stats,
                           const float* __restrict__ gamma,
                           const float* __restrict__ beta,
                           float* __restrict__ dst, int total) {
  int i = blockIdx.x * blockDim.x + threadIdx.x;
  if (i >= total) return;
  int o = (i >> 12) & (OO - 1);
  float y = (pre[i] - stats[o * 2]) * stats[o * 2 + 1] * gamma[o] + beta[o];
  dst[i] = (y >= 0.f) ? y : NEG_SLOPE * y;
}

// ---------------------------------------------------------------------------
extern "C" void kernel_launch(void* const* d_in, const int* in_sizes, int n_in,
                              void* d_out, int out_size, void* d_ws,
                              size_t ws_size, hipStream_t stream) {
  const float* f_in = (const float*)d_in[0];
  const float* w_local = (const float*)d_in[1];
  const float* g_local = (const float*)d_in[2];
  const float* b_local = (const float*)d_in[3];
  const float* w_sem = (const float*)d_in[4];
  const float* g_sem = (const float*)d_in[5];
  const float* b_sem = (const float*)d_in[6];
  const float* w_full = (const float*)d_in[7];
  const float* g_full = (const float*)d_in[8];
  const float* b_full = (const float*)d_in[9];
  const float* wq = (const float*)d_in[10];
  const float* bq = (const float*)d_in[11];
  const float* wk = (const float*)d_in[12];
  const float* bk = (const float*)d_in[13];
  const float* wv = (const float*)d_in[14];
  const float* bv = (const float*)d_in[15];

  float* wsf = (float*)d_ws;
  const size_t SZ_BN = (size_t)BB * NN;            // 16384
  const size_t SZ_BDN = (size_t)BB * DD * NN;      // 1048576
  const size_t SZ_BON = (size_t)BB * OO * NN;      // 2097152
  size_t o_xx = 0;
  size_t o_idx = o_xx + SZ_BN;                     // int region
  size_t o_q = o_idx + SZ_BN * KNN;
  size_t o_k = o_q + SZ_BDN;
  size_t o_v = o_k + SZ_BDN;
  size_t o_fgt = o_v + SZ_BDN;
  size_t o_intra = o_fgt + SZ_BDN;
  size_t o_part = o_intra + SZ_BON;                // 3 x 1024*128*2
  size_t o_stats = o_part + 3 * 1024 * OO * 2;
  size_t o_inter = o_q;  // q,k dead after attention
  size_t o_out = o_v;    // v,fgt dead after inter GEMM

  float* xx = wsf + o_xx;
  int* topk = (int*)(wsf + o_idx);
  float* q = wsf + o_q;
  float* k = wsf + o_k;
  float* v = wsf + o_v;
  float* fgt = wsf + o_fgt;
  float* intra = wsf + o_intra;
  float* inter = wsf + o_inter;
  float* outp = wsf + o_out;
  float* part_i = wsf + o_part;
  float* part_n = part_i + 1024 * OO * 2;
  float* part_o = part_n + 1024 * OO * 2;
  float* st_i = wsf + o_stats;
  float* st_n = st_i + 2 * OO;
  float* st_o = st_n + 2 * OO;

  // 1. squared norms
  k_sq<<<(BB * NN + 255) / 256, 256, 0, stream>>>(f_in, xx);
  // 2. q/k/v projections (WMMA)
  k_proj<<<dim3(BB * NN / 16, 3), 128, 0, stream>>>(f_in, wq, bq, wk, bk, wv,
                                                    bv, q, k, v);
  // 3. Gram + top-16 (WMMA)
  k_topk<<<BB * NN / 16, 128, 0, stream>>>(f_in, xx, topk);
  // 4. attention (WMMA, two-pass streaming softmax)
  k_attn<<<BB * NN / 16 / 8, 256, 0, stream>>>(q, k, v, fgt);
  // 5. local kNN conv + max over k (WMMA) + BN partials
  k_intra<<<BB * NN / 16, 256, 0, stream>>>(f_in, w_local, topk, intra, part_i);
  k_redstats<<<1, 128, 0, stream>>>(part_i, 1024, (float)(BB * NN * KNN), st_i);
  k_finalize<<<(int)(SZ_BON / 256), 256, 0, stream>>>(intra, st_i, g_local,
                                                      b_local, intra,
                                                      (int)SZ_BON);
  // 6. semantic branch GEMM (WMMA) + BN
  k_cgemm<<<BB * NN / 16, 256, 0, stream>>>(w_sem, DD, fgt, DD, fgt, DD, DD,
                                            inter, part_n);
  k_redstats<<<1, 128, 0, stream>>>(part_n, 1024, (float)(BB * NN), st_n);
  k_finalize<<<(int)(SZ_BON / 256), 256, 0, stream>>>(inter, st_n, g_sem, b_sem,
                                                      inter, (int)SZ_BON);
  // 7. fused head GEMM over concat(intra, inter) (WMMA) + BN -> d_out
  k_cgemm<<<BB * NN / 16, 256, 0, stream>>>(w_full, 2 * OO, intra, OO, inter,
                                            OO, OO, outp, part_o);
  k_redstats<<<1, 128, 0, stream>>>(part_o, 1024, (float)(BB * NN), st_o);
  k_finalize<<<(int)(SZ_BON / 256), 256, 0, stream>>>(outp, st_o, g_full,
                                                      b_full, (float*)d_out,
                                                      (int)SZ_BON);
  (void)in_sizes; (void)n_in; (void)out_size; (void)ws_size;
}